// CategoricalDQNModel_43207370998149
// MI455X (gfx1250) — compile-verified
//
#include <hip/hip_runtime.h>
#include <hip/hip_bf16.h>

typedef float v2f __attribute__((ext_vector_type(2)));
typedef float v8f __attribute__((ext_vector_type(8)));

#define NUM_ATOMS   51
#define NUM_ACTIONS 8
__device__ __constant__ float kVMIN = -10.0f;
__device__ __constant__ float kVMAX = 10.0f;
#define DZF      0.4f            /* (10-(-10))/50 */
#define INV_DZF  2.5f
#define DISCOUNTF 0.99f

__device__ __forceinline__ float waveMax(float v) {
#pragma unroll
  for (int m = 16; m > 0; m >>= 1) v = fmaxf(v, __shfl_xor(v, m, 32));
  return v;
}
__device__ __forceinline__ float waveSum(float v) {
#pragma unroll
  for (int m = 16; m > 0; m >>= 1) v += __shfl_xor(v, m, 32);
  return v;
}
// constant-index select chain (avoid dynamic extractelement)
__device__ __forceinline__ float pick8(v8f v, int a) {
  float r = v[0];
  r = (a == 1) ? v[1] : r;
  r = (a == 2) ? v[2] : r;
  r = (a == 3) ? v[3] : r;
  r = (a == 4) ? v[4] : r;
  r = (a == 5) ? v[5] : r;
  r = (a == 6) ? v[6] : r;
  r = (a == 7) ? v[7] : r;
  return r;
}
// gather ls[idx] (idx in [0,50]) from values distributed lane->(idx, idx+32)
__device__ __forceinline__ float gatherLS(float ls_lo, float ls_hi, int idx) {
  float a = __shfl(ls_lo, idx & 31, 32);   // ds_bpermute
  float b = __shfl(ls_hi, idx & 31, 32);
  return (idx < 32) ? a : b;
}
__device__ __forceinline__ float projContrib(int j, float p, float r, float t,
                                             float ls_lo, float ls_hi) {
  float step = -10.0f + (float)j * DZF;
  float tz = r + (1.0f - t) * (step * DISCOUNTF);
  tz = fminf(fmaxf(tz, -10.0f), 10.0f);
  float b = (tz + 10.0f) * INV_DZF;          // in [0, 50]
  float l = floorf(b), u = ceilf(b);
  int li = (int)l, ui = (int)u;
  return p * ((b - l) * gatherLS(ls_lo, ls_hi, li) +
              (u - b) * gatherLS(ls_lo, ls_hi, ui));
}

extern "C" __global__ __launch_bounds__(256)
void c51_loss_kernel(const float* __restrict__ train_logits,
                     const float* __restrict__ target_logits,
                     const float* __restrict__ rewards,
                     const int*   __restrict__ actions,
                     const int*   __restrict__ terminals,
                     float* __restrict__ out, int batch) {
  const int Bm1  = batch - 1;                      // number of outputs
  const int lane = threadIdx.x & 31;
  const long long wave = ((long long)blockIdx.x * blockDim.x + threadIdx.x) >> 5;
  const long long i0 = wave * 2;                   // first output index of this wave
  if (i0 >= Bm1) return;                           // wave-uniform exit

  const int half = lane >> 4;                      // K%4 in {0,1} vs {2,3}
  const int Lm   = lane & 15;                      // matrix row M = inst*8 + action
  const int inst = Lm >> 3;
  const int act  = Lm & 7;

  // ---- early prefetch of the two train rows (action known before phase 1) ----
  {
    long long o1 = i0 + 1; if (o1 > Bm1 - 1) o1 = Bm1 - 1;
    int a0 = actions[i0 + 1];
    int a1 = actions[o1 + 1];
    __builtin_prefetch(train_logits + ((i0 + 1) * NUM_ACTIONS + a0) * NUM_ATOMS, 0, 1);
    __builtin_prefetch(train_logits + ((o1 + 1) * NUM_ACTIONS + a1) * NUM_ATOMS, 0, 1);
  }

  // ---- phase 1: softmax stats + value dot for 16 rows (2 instances x 8 actions)
  long long rowIdx = i0 + 1 + inst;                // target row for this matrix row
  if (rowIdx > Bm1) rowIdx = Bm1;                  // clamp tail (results unused)
  const float* rowp = target_logits + (rowIdx * NUM_ACTIONS + act) * NUM_ATOMS;

  float xs[26];
  const int kb = 2 * half;
#pragma unroll
  for (int c = 0; c < 13; ++c) {
    int k0 = 4 * c + kb;                           // even, <= 50
    xs[2 * c]     = rowp[k0];
    xs[2 * c + 1] = (k0 + 1 < NUM_ATOMS) ? rowp[k0 + 1] : -__builtin_inff();
  }
  float mymax = -__builtin_inff();
#pragma unroll
  for (int q = 0; q < 26; ++q) mymax = fmaxf(mymax, xs[q]);
  float rowmax = fmaxf(mymax, __shfl_xor(mymax, 16, 32));   // per-row max
#pragma unroll
  for (int q = 0; q < 26; ++q) xs[q] = __expf(xs[q] - rowmax); // exp (pad -> 0)

  // 16x51 exp-matrix  x  51x2 [ones | steps]  via 13 accumulating WMMAs
  v8f acc = {};
#pragma unroll
  for (int c = 0; c < 13; ++c) {
    int k0 = 4 * c + kb;
    v2f a; a.x = xs[2 * c]; a.y = xs[2 * c + 1];
    float s0 = -10.0f + (float)k0 * DZF;
    float s1 = -10.0f + (float)(k0 + 1) * DZF;
    v2f b;
    b.x = (Lm == 0) ? 1.0f : ((Lm == 1) ? s0 : 0.0f);
    b.y = (Lm == 0) ? 1.0f : ((Lm == 1) ? s1 : 0.0f);
    acc = __builtin_amdgcn_wmma_f32_16x16x4_f32(false, a, false, b,
                                                (short)0, acc, false, false);
  }
  // D layout: lane0 col0 rows0-7 = sums(inst0); lane1 col1 = numerators(inst0)
  //           lane16/17 = rows 8-15 (inst1)
  v8f nums;
#pragma unroll
  for (int v = 0; v < 8; ++v) nums[v] = __shfl_xor(acc[v], 1, 32);
  int best = 0;
  {
    float bv = nums[0] / acc[0];
#pragma unroll
    for (int a2 = 1; a2 < 8; ++a2) {
      float va = nums[a2] / acc[a2];
      if (va > bv) { bv = va; best = a2; }         // first-max tie-break
    }
  }
  const int A0 = __shfl(best, 0, 32);
  const int A1 = __shfl(best, 16, 32);
  const float mA0 = __shfl(rowmax, A0, 32);        // lane A0 holds row A0 max
  const float mA1 = __shfl(rowmax, 8 + A1, 32);
  const float sA0 = __shfl(pick8(acc, A0), 0, 32);
  const float sA1 = __shfl(pick8(acc, A1), 16, 32);

  // ---- phase 2: projection + cross-entropy, one instance at a time ----
#pragma unroll
  for (int q = 0; q < 2; ++q) {
    long long oi = i0 + q;
    if (oi >= Bm1) break;                          // wave-uniform
    const int   Astar = q ? A1 : A0;
    const float mA    = q ? mA1 : mA0;
    const float sA    = q ? sA1 : sA0;
    const long long row = oi + 1;

    const float* tgt = target_logits + (row * NUM_ACTIONS + Astar) * NUM_ATOMS;
    const int   atr  = actions[row];
    const float* trn = train_logits + (row * NUM_ACTIONS + atr) * NUM_ATOMS;
    const float r  = rewards[oi];
    const float tt = (float)terminals[oi];

    const int j1 = lane, j2 = lane + 32;
    const bool v2ok = (j2 < NUM_ATOMS);

    // log_softmax of selected train row (atoms striped lane, lane+32)
    float tx1 = trn[j1];
    float tx2 = v2ok ? trn[j2] : -__builtin_inff();
    float tmax = waveMax(fmaxf(tx1, tx2));
    float te   = __expf(tx1 - tmax) + (v2ok ? __expf(tx2 - tmax) : 0.0f);
    float logz = __logf(waveSum(te));
    float ls1 = tx1 - tmax - logz;
    float ls2 = tx2 - tmax - logz;                 // only lanes<19 ever gathered

    // selected target probs (reuse row max/sum from WMMA phase)
    float p1 = __expf(tgt[j1] - mA) / sA;
    float p2 = (v2ok ? __expf(tgt[j2] - mA) : 0.0f) / sA;

    // fused projection + dot: sum_j p_j[(b-l)ls[l] + (u-b)ls[u]]
    float contrib = projContrib(j1, p1, r, tt, ls1, ls2)
                  + projContrib(j2, p2, r, tt, ls1, ls2);
    float total = waveSum(contrib);
    if (lane == 0) out[oi] = -total;
  }
}

extern "C" void kernel_launch(void* const* d_in, const int* in_sizes, int n_in,
                              void* d_out, int out_size, void* d_ws, size_t ws_size,
                              hipStream_t stream) {
  const float* train     = (const float*)d_in[0];
  const float* target    = (const float*)d_in[1];
  const float* rewards   = (const float*)d_in[2];
  const int*   actions   = (const int*)d_in[3];
  const int*   terminals = (const int*)d_in[4];
  float* out = (float*)d_out;

  const int batch = in_sizes[2];                   // rewards length == BATCH
  const long long outputs = (long long)batch - 1;
  const long long waves   = (outputs + 1) / 2;     // 2 instances per wave32
  const long long threads = waves * 32;
  const int block = 256;                           // 8 waves / block
  const long long blocks = (threads + block - 1) / block;

  c51_loss_kernel<<<(dim3)(unsigned)blocks, block, 0, stream>>>(
      train, target, rewards, actions, terminals, out, batch);
}